// SemanticAligner_63617055588831
// MI455X (gfx1250) — compile-verified
//
#include <hip/hip_runtime.h>
#include <hip/hip_bf16.h>

#define M_ 3
#define B_ 128
#define T_ 512
#define D_ 1024
#define P_ 256

typedef __attribute__((ext_vector_type(2))) float v2f;
typedef __attribute__((ext_vector_type(8))) float v8f;

// ---------------------------------------------------------------------------
// Kernel 1: partial mean-pool over T.  feats[M,B,T,D] -> part[4, M*B, D]
// Pure bandwidth streamer: 805 MB read total, float4 coalesced, T split 4-way
// across blockIdx.y for more waves in flight.
// ---------------------------------------------------------------------------
__global__ void pool_partial(const float* __restrict__ feats, float* __restrict__ part) {
  const int row = blockIdx.x;            // m*B + b  in [0, 384)
  const int y   = blockIdx.y;            // T-chunk  in [0, 4)
  const int tid = threadIdx.x;           // 0..255, 4 floats each -> covers D=1024
  const float4* base =
      (const float4*)(feats + ((size_t)row * T_ + (size_t)y * (T_ / 4)) * D_);
  float4 acc = {0.f, 0.f, 0.f, 0.f};
#pragma unroll 8
  for (int t = 0; t < T_ / 4; ++t) {
    float4 v = base[(size_t)t * (D_ / 4) + tid];
    acc.x += v.x; acc.y += v.y; acc.z += v.z; acc.w += v.w;
  }
  float4* dst = (float4*)(part + ((size_t)y * (M_ * B_) + row) * D_);
  dst[tid] = acc;
}

// part[4, M*B, D] -> pooled[M*B, D] (scaled by 1/T)
__global__ void pool_finalize(const float* __restrict__ part, float* __restrict__ pooled) {
  const int row = blockIdx.x;
  const int tid = threadIdx.x;
  float4 acc = {0.f, 0.f, 0.f, 0.f};
#pragma unroll
  for (int y = 0; y < 4; ++y) {
    float4 v = ((const float4*)(part + ((size_t)y * (M_ * B_) + row) * D_))[tid];
    acc.x += v.x; acc.y += v.y; acc.z += v.z; acc.w += v.w;
  }
  const float s = 1.0f / (float)T_;
  acc.x *= s; acc.y *= s; acc.z *= s; acc.w *= s;
  ((float4*)(pooled + (size_t)row * D_))[tid] = acc;
}

__device__ __forceinline__ float gelu_exact(float x) {
  // 0.5*x*(1+erf(x/sqrt(2))) -- matches jax.nn.gelu(approximate=False)
  return 0.5f * x * (1.0f + erff(x * 0.70710678118654752440f));
}

// ---------------------------------------------------------------------------
// Kernel 2: h = GELU(pooled @ W1 + b1), fp32 WMMA 16x16x4.
// One wave per 16x16 output tile. grid = 3 * (128/16) * (1024/16) = 1536.
// A-frag: lane=16g+r holds A[r][k+2g], A[r][k+2g+1] (float2, 8B aligned).
// B-frag: lane=16g+n holds B[k+2g][n], B[k+2g+1][n].
// C/D:    VGPR v, lane=16g+n -> row v+8g, col n.
// ---------------------------------------------------------------------------
__global__ void gemm1_gelu(const float* __restrict__ pooled, const float* __restrict__ W1,
                           const float* __restrict__ b1, float* __restrict__ h) {
  const int tile = blockIdx.x;
  const int ct = tile & 63;          // col tile over D
  const int rt = (tile >> 6) & 7;    // row tile over B
  const int m  = tile >> 9;          // modality
  const int lane = threadIdx.x;
  const int g   = lane >> 4;
  const int l16 = lane & 15;

  const float* arow = pooled + (size_t)(m * B_ + rt * 16 + l16) * D_;
  const float* bcol = W1 + (size_t)m * D_ * D_ + ct * 16 + l16;

  v8f c = {};
  for (int k = 0; k < D_; k += 4) {
    v2f a = *(const v2f*)(arow + k + 2 * g);
    v2f b;
    b.x = bcol[(size_t)(k + 2 * g) * D_];
    b.y = bcol[(size_t)(k + 2 * g + 1) * D_];
    c = __builtin_amdgcn_wmma_f32_16x16x4_f32(false, a, false, b, (short)0, c,
                                              false, false);
  }
  const int col = ct * 16 + l16;
  const float bias = b1[m * D_ + col];
#pragma unroll
  for (int v = 0; v < 8; ++v) {
    const int row = rt * 16 + v + 8 * g;
    h[(size_t)(m * B_ + row) * D_ + col] = gelu_exact(c[v] + bias);
  }
}

// ---------------------------------------------------------------------------
// Kernel 3: z = h @ W2 + b2, then LayerNorm(P) then L2-normalize, fused.
// One block (8 waves) per (m, 16-row block): each wave computes 2 col-tiles,
// z staged in LDS, row reductions block-local. grid = 3*8 = 24 blocks.
// ---------------------------------------------------------------------------
__global__ void gemm2_norm(const float* __restrict__ h, const float* __restrict__ W2,
                           const float* __restrict__ b2, const float* __restrict__ gamma,
                           const float* __restrict__ beta, float* __restrict__ out) {
  const int m   = blockIdx.x >> 3;
  const int rb  = blockIdx.x & 7;
  const int tid = threadIdx.x;       // 0..255
  const int wave = tid >> 5;
  const int lane = tid & 31;
  const int g   = lane >> 4;
  const int l16 = lane & 15;

  __shared__ float s_z[16][P_ + 1];
  __shared__ float s_r1[256];
  __shared__ float s_r2[256];
  __shared__ float s_mean[16], s_rstd[16], s_inv[16];

  const float* arow  = h + (size_t)(m * B_ + rb * 16 + l16) * D_;
  const float* bbase = W2 + (size_t)m * D_ * P_ + l16;
  const int ct0 = wave * 2, ct1 = wave * 2 + 1;
  const float* bc0 = bbase + ct0 * 16;
  const float* bc1 = bbase + ct1 * 16;

  v8f c0 = {}, c1 = {};
  for (int k = 0; k < D_; k += 4) {
    v2f a = *(const v2f*)(arow + k + 2 * g);
    v2f x0, x1;
    x0.x = bc0[(size_t)(k + 2 * g) * P_];
    x0.y = bc0[(size_t)(k + 2 * g + 1) * P_];
    x1.x = bc1[(size_t)(k + 2 * g) * P_];
    x1.y = bc1[(size_t)(k + 2 * g + 1) * P_];
    c0 = __builtin_amdgcn_wmma_f32_16x16x4_f32(false, a, false, x0, (short)0, c0,
                                               false, false);
    c1 = __builtin_amdgcn_wmma_f32_16x16x4_f32(false, a, false, x1, (short)0, c1,
                                               false, false);
  }
  {
    const int col0 = ct0 * 16 + l16, col1 = ct1 * 16 + l16;
    const float bia0 = b2[m * P_ + col0], bia1 = b2[m * P_ + col1];
#pragma unroll
    for (int v = 0; v < 8; ++v) {
      const int r = v + 8 * g;
      s_z[r][col0] = c0[v] + bia0;
      s_z[r][col1] = c1[v] + bia1;
    }
  }
  __syncthreads();

  // per-row mean / var over P=256 (16 threads per row, 16 elems per thread)
  const int r   = tid >> 4;
  const int seg = tid & 15;
  float s = 0.f, sq = 0.f;
#pragma unroll
  for (int i = 0; i < 16; ++i) {
    const float z = s_z[r][seg * 16 + i];
    s += z; sq += z * z;
  }
  s_r1[tid] = s; s_r2[tid] = sq;
  __syncthreads();
  if (tid < 16) {
    float su = 0.f, sqs = 0.f;
#pragma unroll
    for (int j = 0; j < 16; ++j) { su += s_r1[tid * 16 + j]; sqs += s_r2[tid * 16 + j]; }
    const float mu  = su * (1.0f / P_);
    const float var = sqs * (1.0f / P_) - mu * mu;
    s_mean[tid] = mu;
    s_rstd[tid] = rsqrtf(var + 1e-5f);
  }
  __syncthreads();

  // LayerNorm affine + accumulate y^2 for L2 norm
  const float mu = s_mean[r], rstd = s_rstd[r];
  float nsq = 0.f;
#pragma unroll
  for (int i = 0; i < 16; ++i) {
    const int cc = seg * 16 + i;
    const float yv = (s_z[r][cc] - mu) * rstd * gamma[m * P_ + cc] + beta[m * P_ + cc];
    s_z[r][cc] = yv;
    nsq += yv * yv;
  }
  s_r1[tid] = nsq;
  __syncthreads();
  if (tid < 16) {
    float su = 0.f;
#pragma unroll
    for (int j = 0; j < 16; ++j) su += s_r1[tid * 16 + j];
    s_inv[tid] = 1.0f / fmaxf(sqrtf(su), 1e-12f);
  }
  __syncthreads();

  const float inv = s_inv[r];
  float* orow = out + (size_t)(m * B_ + rb * 16 + r) * P_;
#pragma unroll
  for (int i = 0; i < 16; ++i) {
    const int cc = seg * 16 + i;
    orow[cc] = s_z[r][cc] * inv;
  }
}

extern "C" void kernel_launch(void* const* d_in, const int* in_sizes, int n_in,
                              void* d_out, int out_size, void* d_ws, size_t ws_size,
                              hipStream_t stream) {
  const float* feats = (const float*)d_in[0];
  const float* W1    = (const float*)d_in[1];
  const float* b1    = (const float*)d_in[2];
  const float* W2    = (const float*)d_in[3];
  const float* b2    = (const float*)d_in[4];
  const float* gamma = (const float*)d_in[5];
  const float* beta  = (const float*)d_in[6];
  float* out = (float*)d_out;
  float* ws  = (float*)d_ws;

  float* part   = ws;                               // 4 * 384 * 1024 floats (6.3 MB)
  float* pooled = part + (size_t)4 * (M_ * B_) * D_; // 384 * 1024 floats (1.6 MB)
  float* h      = pooled + (size_t)(M_ * B_) * D_;   // 384 * 1024 floats (1.6 MB)

  dim3 g1(M_ * B_, 4);
  pool_partial<<<g1, 256, 0, stream>>>(feats, part);
  pool_finalize<<<M_ * B_, 256, 0, stream>>>(part, pooled);
  gemm1_gelu<<<M_ * (B_ / 16) * (D_ / 16), 32, 0, stream>>>(pooled, W1, b1, h);
  gemm2_norm<<<M_ * (B_ / 16), 256, 0, stream>>>(h, W2, b2, gamma, beta, out);
}